// MLP_81037442941198
// MI455X (gfx1250) — compile-verified
//
#include <hip/hip_runtime.h>
#include <cmath>

// ---------------------------------------------------------------------------
// MLP 3->10->20->25->15->10->5->3 with training-mode BN + tanh between layers.
// Stage kernel: BN+tanh(prev raw h) -> GEMM (V_WMMA_F32_16X16X4_F32)
//               -> store raw h -> accumulate per-feature sum / sumsq.
// Internal buffers padded to 16/32 columns => unguarded coalesced stores with
// immediate offsets; stats via register accumulation + shfl_xor(16) + atomics.
// ---------------------------------------------------------------------------

typedef float v2f __attribute__((ext_vector_type(2)));
typedef float v8f __attribute__((ext_vector_type(8)));

#define BN_EPS 1e-5f

__device__ __forceinline__ float dtanh(float x) {
#if __has_builtin(__builtin_amdgcn_tanhf)
  return __builtin_amdgcn_tanhf(x);   // gfx1250 v_tanh_f32 trans op
#else
  return tanhf(x);
#endif
}

// D = A(16x4 f32) * B(4x16 f32) + C(16x16 f32), full fp32 accumulate.
__device__ __forceinline__ v8f wmma4(v2f a, v2f b, v8f c) {
  return __builtin_amdgcn_wmma_f32_16x16x4_f32(
      /*neg_a=*/false, a, /*neg_b=*/false, b,
      /*c_mod=*/(short)0, c, /*reuse_a=*/false, /*reuse_b=*/false);
}

__global__ void zero_stats_kernel(float* __restrict__ p, int n) {
  int i = blockIdx.x * blockDim.x + threadIdx.x;
  if (i < n) p[i] = 0.f;
}

// FIN:       actual input feature count of this linear layer
// STRIDE_IN: row stride (floats) of input buffer (16/32 internal, 3 for x)
// FOUT:      actual output feature count
// FOUTPAD:   row stride of output buffer (multiple of 16; ignored when LAST)
// FIRST:     input is x (no BN on input)
// LAST:      output goes to d_out (stride FOUT=3, guarded stores, no stats)
template <int FIN, int STRIDE_IN, int FOUT, int FOUTPAD, bool FIRST, bool LAST>
__global__ __launch_bounds__(256) void mlp_stage(
    const float* __restrict__ in, const float* __restrict__ W,
    const float* __restrict__ bias, const float* __restrict__ gamma,
    const float* __restrict__ beta, const float* __restrict__ statsIn,
    float* __restrict__ out, float* __restrict__ statsOut, int nrows) {
  constexpr int KPAD = ((FIN + 3) / 4) * 4;  // K padded to wmma K=4 chunks
  constexpr int NCH  = KPAD / 4;             // number of K chunks
  constexpr int NT   = LAST ? 1 : (FOUTPAD / 16);  // N tiles of 16 columns
  static_assert(LAST || (FOUTPAD % 16 == 0), "padded stores must be unguarded");

  const int lane  = threadIdx.x & 31;
  const int half  = lane >> 4;   // 0: lanes 0-15, 1: lanes 16-31
  const int lcol  = lane & 15;
  const int wave  = (blockIdx.x * blockDim.x + threadIdx.x) >> 5;
  const int nwave = (gridDim.x * blockDim.x) >> 5;
  const float invN = 1.0f / (float)nrows;

  // BN affine for the input columns this lane touches in its A fragments:
  // chunk c, element e -> column k = 4c + 2*half + e.
  float nsc[NCH][2], nsh[NCH][2];
  if (!FIRST) {
#pragma unroll
    for (int c = 0; c < NCH; ++c) {
#pragma unroll
      for (int e = 0; e < 2; ++e) {
        const int k  = c * 4 + half * 2 + e;
        const float mean = statsIn[k] * invN;
        const float var  = fmaf(-mean, mean, statsIn[32 + k] * invN);
        const float istd = rsqrtf(var + BN_EPS);
        const float g  = (k < FIN) ? gamma[k] : 0.f;
        const float bt = (k < FIN) ? beta[k] : 0.f;
        nsc[c][e] = istd * g;
        nsh[c][e] = fmaf(-mean, nsc[c][e], bt);
      }
    }
  }

  // Loop-invariant B fragments (weights) and bias splat per N-tile.
  // B 4x16 layout mirrors A: lanes 0-15 hold K=k0..k0+1, lanes 16-31 K=k0+2..3.
  v2f Bf[NT][NCH];
  float bv[NT];
#pragma unroll
  for (int nt = 0; nt < NT; ++nt) {
    const int col = nt * 16 + lcol;
    bv[nt] = (col < FOUT) ? bias[col] : 0.f;
#pragma unroll
    for (int c = 0; c < NCH; ++c) {
#pragma unroll
      for (int e = 0; e < 2; ++e) {
        const int k = c * 4 + half * 2 + e;
        Bf[nt][c][e] = (k < FIN && col < FOUT) ? W[k * FOUT + col] : 0.f;
      }
    }
  }

  float s_sum[NT], s_ssq[NT];
#pragma unroll
  for (int nt = 0; nt < NT; ++nt) { s_sum[nt] = 0.f; s_ssq[nt] = 0.f; }

  const int ntiles = nrows >> 4;  // 16 rows per wave-tile (2M % 16 == 0)
  for (int t = wave; t < ntiles; t += nwave) {
    const int row0 = t << 4;
    const int arow = row0 + lcol;              // A-matrix row for this lane
    const float* ip = in + (size_t)arow * STRIDE_IN;  // one base, imm offsets

    v8f acc[NT];
#pragma unroll
    for (int nt = 0; nt < NT; ++nt)
#pragma unroll
      for (int j = 0; j < 8; ++j) acc[nt][j] = bv[nt];

#pragma unroll
    for (int c = 0; c < NCH; ++c) {
      const int kb = c * 4 + half * 2;
      v2f A;
      if (FIRST) {
        // FIN==3: k = {0,1} (half 0) or {2,3->pad} (half 1). Branchless:
        // always load the valid col (half*2) and col 1; select 0 for k==3.
        A[0] = ip[half * 2];
        const float t1 = ip[1];
        A[1] = half ? 0.f : t1;
      } else {
        // STRIDE_IN even, kb even -> 8B-aligned b64 load at imm offset.
        const float2 hv = *(const float2*)(ip + kb);
        A[0] = dtanh(fmaf(hv.x, nsc[c][0], nsh[c][0]));
        A[1] = dtanh(fmaf(hv.y, nsc[c][1], nsh[c][1]));
      }
#pragma unroll
      for (int nt = 0; nt < NT; ++nt) acc[nt] = wmma4(A, Bf[nt][c], acc[nt]);
    }

    // D layout: VGPR j -> row (row0 + j + 8*half), col = nt*16 + (lane&15).
    if (LAST) {
      const int col = lcol;
      if (col < FOUT) {
        float* op = out + (size_t)(row0 + half * 8) * FOUT + col;
#pragma unroll
        for (int j = 0; j < 8; ++j) op[j * FOUT] = acc[0][j];
      }
    } else {
#pragma unroll
      for (int nt = 0; nt < NT; ++nt) {
        float* op = out + (size_t)(row0 + half * 8) * FOUTPAD + nt * 16 + lcol;
#pragma unroll
        for (int j = 0; j < 8; ++j) {
          const float v = acc[nt][j];
          op[j * FOUTPAD] = v;               // contiguous 64B per half-wave
          s_sum[nt] += v;
          s_ssq[nt] = fmaf(v, v, s_ssq[nt]);
        }
      }
    }
  }

  if (!LAST) {
    // Combine the two half-wave row groups (same column set), then one
    // global atomic per (lane<16, valid column).
#pragma unroll
    for (int nt = 0; nt < NT; ++nt) {
      const float s = s_sum[nt] + __shfl_xor(s_sum[nt], 16, 32);
      const float q = s_ssq[nt] + __shfl_xor(s_ssq[nt], 16, 32);
      const int col = nt * 16 + lcol;
      if (half == 0 && col < FOUT) {
        atomicAdd(&statsOut[col], s);
        atomicAdd(&statsOut[32 + col], q);
      }
    }
  }
}

extern "C" void kernel_launch(void* const* d_in, const int* in_sizes, int n_in,
                              void* d_out, int out_size, void* d_ws,
                              size_t ws_size, hipStream_t stream) {
  (void)n_in; (void)out_size; (void)ws_size;
  // dict order: x, W0..W6, b0..b6, gamma0..gamma5, beta0..beta5
  const float* x = (const float*)d_in[0];
  const float* W[7];  for (int i = 0; i < 7; ++i) W[i] = (const float*)d_in[1 + i];
  const float* b[7];  for (int i = 0; i < 7; ++i) b[i] = (const float*)d_in[8 + i];
  const float* g[6];  for (int i = 0; i < 6; ++i) g[i] = (const float*)d_in[15 + i];
  const float* be[6]; for (int i = 0; i < 6; ++i) be[i] = (const float*)d_in[21 + i];
  const int n = in_sizes[0] / 3;  // 2,000,000 rows

  // ws layout: bufA (n*32 f32: h0/h2/h4), bufB (n*32 f32: h1/h3/h5),
  // stats: 6 layers * 64 floats (32 sums + 32 sumsq each).
  float* ws   = (float*)d_ws;
  float* bufA = ws;
  float* bufB = ws + (size_t)n * 32;
  float* st   = ws + (size_t)n * 64;

  zero_stats_kernel<<<2, 256, 0, stream>>>(st, 6 * 64);

  dim3 blk(256), grd(1024);
  // <FIN, STRIDE_IN, FOUT, FOUTPAD, FIRST, LAST>
  mlp_stage<3, 3, 10, 16, true, false><<<grd, blk, 0, stream>>>(
      x, W[0], b[0], nullptr, nullptr, nullptr, bufA, st + 0 * 64, n);
  mlp_stage<10, 16, 20, 32, false, false><<<grd, blk, 0, stream>>>(
      bufA, W[1], b[1], g[0], be[0], st + 0 * 64, bufB, st + 1 * 64, n);
  mlp_stage<20, 32, 25, 32, false, false><<<grd, blk, 0, stream>>>(
      bufB, W[2], b[2], g[1], be[1], st + 1 * 64, bufA, st + 2 * 64, n);
  mlp_stage<25, 32, 15, 16, false, false><<<grd, blk, 0, stream>>>(
      bufA, W[3], b[3], g[2], be[2], st + 2 * 64, bufB, st + 3 * 64, n);
  mlp_stage<15, 16, 10, 16, false, false><<<grd, blk, 0, stream>>>(
      bufB, W[4], b[4], g[3], be[3], st + 3 * 64, bufA, st + 4 * 64, n);
  mlp_stage<10, 16, 5, 16, false, false><<<grd, blk, 0, stream>>>(
      bufA, W[5], b[5], g[4], be[4], st + 4 * 64, bufB, st + 5 * 64, n);
  mlp_stage<5, 16, 3, 16, false, true><<<grd, blk, 0, stream>>>(
      bufB, W[6], b[6], g[5], be[5], st + 5 * 64, (float*)d_out, nullptr, n);
}